// ContrastiveLoss_9070970929207
// MI455X (gfx1250) — compile-verified
//
#include <hip/hip_runtime.h>
#include <math.h>

typedef float v2f __attribute__((ext_vector_type(2)));
typedef float v8f __attribute__((ext_vector_type(8)));

#define WMMA_F32(a, b, c) \
  __builtin_amdgcn_wmma_f32_16x16x4_f32(false, (a), false, (b), (short)0, (c), false, false)

__device__ __forceinline__ v2f ldv2(const float* p) { return *(const v2f*)p; }

// ---------------------------------------------------------------------------
// C[M,N] = X[M,K] @ Y[N,K]^T   (NT GEMM, f32 WMMA 16x16x4)
// one wave per block computing a 32(M) x 64(N) tile; software-pipelined.
// grid = (M/32, N/64). K % 4 == 0, K > 4. All row-major.
// ---------------------------------------------------------------------------
__global__ __launch_bounds__(32) void gemm_nt_kernel(
    const float* __restrict__ X, const float* __restrict__ Y,
    float* __restrict__ C, int K, int ldc)
{
  const int lane = threadIdx.x & 31;
  const int h = lane >> 4;      // K-pair half select
  const int ml = lane & 15;     // row/col within tile
  const long row0 = (long)blockIdx.x * 32;
  const long col0 = (long)blockIdx.y * 64;

  const float* xp0 = X + (row0 +  0 + ml) * (long)K + 2 * h;
  const float* xp1 = X + (row0 + 16 + ml) * (long)K + 2 * h;
  const float* yp0 = Y + (col0 +  0 + ml) * (long)K + 2 * h;
  const float* yp1 = Y + (col0 + 16 + ml) * (long)K + 2 * h;
  const float* yp2 = Y + (col0 + 32 + ml) * (long)K + 2 * h;
  const float* yp3 = Y + (col0 + 48 + ml) * (long)K + 2 * h;

  v8f c00 = {}, c01 = {}, c02 = {}, c03 = {};
  v8f c10 = {}, c11 = {}, c12 = {}, c13 = {};

  // prologue fragments
  v2f a0 = ldv2(xp0), a1 = ldv2(xp1);
  v2f b0 = ldv2(yp0), b1 = ldv2(yp1), b2 = ldv2(yp2), b3 = ldv2(yp3);

  for (int k0 = 0; k0 < K - 4; k0 += 4) {
    // issue next iteration's loads before consuming current fragments
    v2f a0n = ldv2(xp0 + k0 + 4), a1n = ldv2(xp1 + k0 + 4);
    v2f b0n = ldv2(yp0 + k0 + 4), b1n = ldv2(yp1 + k0 + 4);
    v2f b2n = ldv2(yp2 + k0 + 4), b3n = ldv2(yp3 + k0 + 4);

    c00 = WMMA_F32(a0, b0, c00);
    c01 = WMMA_F32(a0, b1, c01);
    c02 = WMMA_F32(a0, b2, c02);
    c03 = WMMA_F32(a0, b3, c03);
    c10 = WMMA_F32(a1, b0, c10);
    c11 = WMMA_F32(a1, b1, c11);
    c12 = WMMA_F32(a1, b2, c12);
    c13 = WMMA_F32(a1, b3, c13);

    a0 = a0n; a1 = a1n; b0 = b0n; b1 = b1n; b2 = b2n; b3 = b3n;
  }
  c00 = WMMA_F32(a0, b0, c00);
  c01 = WMMA_F32(a0, b1, c01);
  c02 = WMMA_F32(a0, b2, c02);
  c03 = WMMA_F32(a0, b3, c03);
  c10 = WMMA_F32(a1, b0, c10);
  c11 = WMMA_F32(a1, b1, c11);
  c12 = WMMA_F32(a1, b2, c12);
  c13 = WMMA_F32(a1, b3, c13);

  float* cp0 = C + (row0 +  0 + 8 * h) * (long)ldc + col0 + ml;
  float* cp1 = C + (row0 + 16 + 8 * h) * (long)ldc + col0 + ml;
  #pragma unroll
  for (int r = 0; r < 8; ++r) {
    cp0[(long)r * ldc +  0] = c00[r];
    cp0[(long)r * ldc + 16] = c01[r];
    cp0[(long)r * ldc + 32] = c02[r];
    cp0[(long)r * ldc + 48] = c03[r];
    cp1[(long)r * ldc +  0] = c10[r];
    cp1[(long)r * ldc + 16] = c11[r];
    cp1[(long)r * ldc + 32] = c12[r];
    cp1[(long)r * ldc + 48] = c13[r];
  }
}

// ---------------------------------------------------------------------------
// C[M,N] = X[M,K] @ Y[K,N]     (NN GEMM, f32 WMMA 16x16x4)
// grid = (M/32, N/64), one wave per block; software-pipelined.
// ---------------------------------------------------------------------------
__global__ __launch_bounds__(32) void gemm_nn_kernel(
    const float* __restrict__ X, const float* __restrict__ Y,
    float* __restrict__ C, int K, int N)
{
  const int lane = threadIdx.x & 31;
  const int h = lane >> 4;
  const int ml = lane & 15;
  const long row0 = (long)blockIdx.x * 32;
  const long col0 = (long)blockIdx.y * 64;

  const float* xp0 = X + (row0 +  0 + ml) * (long)K + 2 * h;
  const float* xp1 = X + (row0 + 16 + ml) * (long)K + 2 * h;

  auto ldb = [&](int k0, int off) -> v2f {
    const float* yr = Y + (long)(k0 + 2 * h) * N + col0 + ml + off;
    v2f b; b.x = yr[0]; b.y = yr[(long)N];
    return b;
  };

  v8f c00 = {}, c01 = {}, c02 = {}, c03 = {};
  v8f c10 = {}, c11 = {}, c12 = {}, c13 = {};

  v2f a0 = ldv2(xp0), a1 = ldv2(xp1);
  v2f b0 = ldb(0, 0), b1 = ldb(0, 16), b2 = ldb(0, 32), b3 = ldb(0, 48);

  for (int k0 = 0; k0 < K - 4; k0 += 4) {
    v2f a0n = ldv2(xp0 + k0 + 4), a1n = ldv2(xp1 + k0 + 4);
    v2f b0n = ldb(k0 + 4, 0),  b1n = ldb(k0 + 4, 16);
    v2f b2n = ldb(k0 + 4, 32), b3n = ldb(k0 + 4, 48);

    c00 = WMMA_F32(a0, b0, c00);
    c01 = WMMA_F32(a0, b1, c01);
    c02 = WMMA_F32(a0, b2, c02);
    c03 = WMMA_F32(a0, b3, c03);
    c10 = WMMA_F32(a1, b0, c10);
    c11 = WMMA_F32(a1, b1, c11);
    c12 = WMMA_F32(a1, b2, c12);
    c13 = WMMA_F32(a1, b3, c13);

    a0 = a0n; a1 = a1n; b0 = b0n; b1 = b1n; b2 = b2n; b3 = b3n;
  }
  c00 = WMMA_F32(a0, b0, c00);
  c01 = WMMA_F32(a0, b1, c01);
  c02 = WMMA_F32(a0, b2, c02);
  c03 = WMMA_F32(a0, b3, c03);
  c10 = WMMA_F32(a1, b0, c10);
  c11 = WMMA_F32(a1, b1, c11);
  c12 = WMMA_F32(a1, b2, c12);
  c13 = WMMA_F32(a1, b3, c13);

  float* cp0 = C + (row0 +  0 + 8 * h) * (long)N + col0 + ml;
  float* cp1 = C + (row0 + 16 + 8 * h) * (long)N + col0 + ml;
  #pragma unroll
  for (int r = 0; r < 8; ++r) {
    cp0[(long)r * N +  0] = c00[r];
    cp0[(long)r * N + 16] = c01[r];
    cp0[(long)r * N + 32] = c02[r];
    cp0[(long)r * N + 48] = c03[r];
    cp1[(long)r * N +  0] = c10[r];
    cp1[(long)r * N + 16] = c11[r];
    cp1[(long)r * N + 32] = c12[r];
    cp1[(long)r * N + 48] = c13[r];
  }
}

// ---------------------------------------------------------------------------
// Fused frame_word kernel: one workgroup per (t,v).
// 5 waves (160 threads). B-matrix (frame[v], 64x768) is staged chunk-wise
// into LDS with CDNA5 async global->LDS copies (ASYNCcnt), then
// S[80,64] = WL[t] @ frame[v]^T via WMMA (A frags from global, B frags from
// LDS), followed by the softmax / weight-mix / bilinear epilogue -> fw[t,v].
// ---------------------------------------------------------------------------
__global__ __launch_bounds__(160) void frame_word_kernel(
    const float* __restrict__ WL, const float* __restrict__ frame,
    const float* __restrict__ Wm,  const float* __restrict__ Fm,
    const float* __restrict__ Fm2, const float* __restrict__ Wm2,
    float* __restrict__ fw)
{
  const int t = blockIdx.y;
  const int v = blockIdx.x;

  __shared__ float S[80 * 68];    // S tile, stride 68 (conflict-free columns)
  __shared__ float P[80 * 68];    // sw (stage 1-2) then sf (stage 3-4)
  __shared__ float Bst[64 * 68];  // staged B chunk: 64 rows x 64 K values
  __shared__ float wl[64];        // word_level[f]
  __shared__ float fl[80];        // frame_level[w]
  __shared__ float stat[4];       // m_wl, s_wl, m_fl, s_fl
  __shared__ float acc[2];        // sent2frame, video2word

  const int tid  = threadIdx.x;
  const int wave = tid >> 5;
  const int lane = tid & 31;
  const int h  = lane >> 4;
  const int ml = lane & 15;

  // ---- GEMM: each wave computes rows [16*wave, 16*wave+16) of S.
  {
    const float* xp = WL + ((long)t * 80 + wave * 16 + ml) * 768 + 2 * h;
    const float* fbase = frame + (long)v * 64 * 768;

    v8f c0 = {}, c1 = {}, c2 = {}, c3 = {};

    for (int k0 = 0; k0 < 768; k0 += 64) {
      // --- async stage: frame[v*64+row][k0 .. k0+63] -> Bst[row][0..63]
      // 64 rows x 16 float4-groups = 1024 async b128 copies, split over 160 thr
      for (int g = tid; g < 1024; g += 160) {
        const int row = g >> 4;
        const int kk  = (g & 15) * 4;
        unsigned lds_off =
            (unsigned)(uintptr_t)(const void*)&Bst[row * 68 + kk];
        unsigned long long gaddr =
            (unsigned long long)(uintptr_t)(fbase + (long)row * 768 + k0 + kk);
        asm volatile("global_load_async_to_lds_b128 %0, %1, off"
                     :: "v"(lds_off), "v"(gaddr)
                     : "memory");
      }
      asm volatile("s_wait_asynccnt 0x0" ::: "memory");
      __syncthreads();

      // --- consume chunk: 16 K-steps of 4, B frags from LDS
      #pragma unroll
      for (int ks = 0; ks < 64; ks += 4) {
        v2f a  = ldv2(xp + k0 + ks);
        v2f b0 = *(const v2f*)&Bst[( 0 + ml) * 68 + ks + 2 * h];
        v2f b1 = *(const v2f*)&Bst[(16 + ml) * 68 + ks + 2 * h];
        v2f b2 = *(const v2f*)&Bst[(32 + ml) * 68 + ks + 2 * h];
        v2f b3 = *(const v2f*)&Bst[(48 + ml) * 68 + ks + 2 * h];
        c0 = WMMA_F32(a, b0, c0);
        c1 = WMMA_F32(a, b1, c1);
        c2 = WMMA_F32(a, b2, c2);
        c3 = WMMA_F32(a, b3, c3);
      }
      __syncthreads();   // protect Bst before next chunk overwrites it
    }

    const int rbase = wave * 16 + 8 * h;
    #pragma unroll
    for (int r = 0; r < 8; ++r) {
      S[(rbase + r) * 68 + ml +  0] = c0[r];
      S[(rbase + r) * 68 + ml + 16] = c1[r];
      S[(rbase + r) * 68 + ml + 32] = c2[r];
      S[(rbase + r) * 68 + ml + 48] = c3[r];
    }
  }
  __syncthreads();

  // ---- Stage 1: column softmax over w (per f): sw -> P
  if (tid < 64) {
    const int f = tid;
    float m = -3.4e38f;
    for (int w = 0; w < 80; ++w) m = fmaxf(m, S[w * 68 + f]);
    float s = 0.f;
    for (int w = 0; w < 80; ++w) {
      float e = __expf((S[w * 68 + f] - m) * 100.0f);
      P[w * 68 + f] = e;
      s += e;
    }
    float inv = 1.0f / s;
    for (int w = 0; w < 80; ++w) P[w * 68 + f] *= inv;
  }
  if (tid < 64) wl[tid] = 0.f;
  if (tid < 80) fl[tid] = 0.f;
  __syncthreads();

  // ---- Stage 2: word_level[f] = sum_u S[u,f] * (sum_w Wm[w,u]*sw[w,f])
  for (int idx = tid; idx < 80 * 64; idx += 160) {
    const int u = idx >> 6;
    const int f = idx & 63;
    float t1 = 0.f;
    for (int w = 0; w < 80; ++w) t1 += Wm[w * 80 + u] * P[w * 68 + f];
    atomicAdd(&wl[f], S[u * 68 + f] * t1);
  }
  __syncthreads();

  // ---- Stage 3: row softmax over f (per w): sf -> P
  if (tid < 80) {
    const int w = tid;
    float m = -3.4e38f;
    for (int f = 0; f < 64; ++f) m = fmaxf(m, S[w * 68 + f]);
    float s = 0.f;
    for (int f = 0; f < 64; ++f) {
      float e = __expf((S[w * 68 + f] - m) * 100.0f);
      P[w * 68 + f] = e;
      s += e;
    }
    float inv = 1.0f / s;
    for (int f = 0; f < 64; ++f) P[w * 68 + f] *= inv;
  }
  __syncthreads();

  // ---- Stage 4: frame_level[w] = sum_u S[w,u] * (sum_f sf[w,f]*Fm[f,u])
  for (int idx = tid; idx < 80 * 64; idx += 160) {
    const int w = idx >> 6;
    const int u = idx & 63;
    float t2 = 0.f;
    for (int f = 0; f < 64; ++f) t2 += P[w * 68 + f] * Fm[f * 64 + u];
    atomicAdd(&fl[w], S[w * 68 + u] * t2);
  }
  __syncthreads();

  // ---- Stage 5 prep: softmax stats for wl (64) and fl (80)
  if (tid == 0) {
    float m = -3.4e38f;
    for (int f = 0; f < 64; ++f) m = fmaxf(m, wl[f]);
    float s = 0.f;
    for (int f = 0; f < 64; ++f) s += __expf((wl[f] - m) * 100.0f);
    stat[0] = m; stat[1] = s; acc[0] = 0.f;
  }
  if (tid == 32) {
    float m = -3.4e38f;
    for (int w = 0; w < 80; ++w) m = fmaxf(m, fl[w]);
    float s = 0.f;
    for (int w = 0; w < 80; ++w) s += __expf((fl[w] - m) * 100.0f);
    stat[2] = m; stat[3] = s; acc[1] = 0.f;
  }
  __syncthreads();

  // ---- Stage 6: bilinear reductions
  if (tid < 64) {
    const int f = tid;
    float e = __expf((wl[f] - stat[0]) * 100.0f) / stat[1];
    float inner = 0.f;
    for (int f2 = 0; f2 < 64; ++f2) inner += Fm2[f * 64 + f2] * wl[f2];
    atomicAdd(&acc[0], e * inner);           // sent2frame
  } else if (tid < 144) {
    const int w = tid - 64;
    float e = __expf((fl[w] - stat[2]) * 100.0f) / stat[3];
    float inner = 0.f;
    for (int u = 0; u < 80; ++u) inner += Wm2[w * 80 + u] * fl[u];
    atomicAdd(&acc[1], e * inner);           // video2word
  }
  __syncthreads();

  if (tid == 0) fw[(long)t * 128 + v] = 0.5f * (acc[0] + acc[1]);
}

// ---------------------------------------------------------------------------
// video_word_sim[t,v] = sum_w softmax_w(A[t,:,v]*100)[w] * sum_u Wl[w,u]*A[t,u,v]
// Am is [10240,128] = [t*80+w, v]. One block per t, thread = v.
// ---------------------------------------------------------------------------
__global__ __launch_bounds__(128) void reduce_video_word_kernel(
    const float* __restrict__ Am, const float* __restrict__ Wl,
    float* __restrict__ out)
{
  const int t = blockIdx.x;
  const int tid = threadIdx.x;
  __shared__ float a[80 * 132];

  for (int idx = tid; idx < 80 * 128; idx += 128) {
    const int w = idx >> 7;
    const int vv = idx & 127;
    a[w * 132 + vv] = Am[((long)t * 80 + w) * 128 + vv];
  }
  __syncthreads();

  const int v = tid;
  float m = -3.4e38f;
  for (int w = 0; w < 80; ++w) m = fmaxf(m, a[w * 132 + v]);
  float s = 0.f;
  for (int w = 0; w < 80; ++w) s += __expf((a[w * 132 + v] - m) * 100.0f);
  const float inv = 1.0f / s;

  float accv = 0.f;
  for (int w = 0; w < 80; ++w) {
    float e = __expf((a[w * 132 + v] - m) * 100.0f) * inv;
    float g = 0.f;
    for (int u = 0; u < 80; ++u) g += Wl[w * 80 + u] * a[u * 132 + v];
    accv += e * g;
  }
  out[(long)t * 128 + v] = accv;
}

// ---------------------------------------------------------------------------
// sentence_frame_sim[t,v] = sum_f' softmax_f(B[t,v,:]*100)[f'] * sum_f Fl[f',f]*B[t,v,f]
// Bm is [128, 8192] = [t, v*64+f]. One block per t, thread = v.
// ---------------------------------------------------------------------------
__global__ __launch_bounds__(128) void reduce_sentence_frame_kernel(
    const float* __restrict__ Bm, const float* __restrict__ Fl,
    float* __restrict__ out)
{
  const int t = blockIdx.x;
  const int tid = threadIdx.x;
  __shared__ float b[64 * 130];   // transposed: b[f][v]

  for (int idx = tid; idx < 8192; idx += 128) {
    const int vv = idx >> 6;
    const int f = idx & 63;
    b[f * 130 + vv] = Bm[(long)t * 8192 + idx];
  }
  __syncthreads();

  const int v = tid;
  float m = -3.4e38f;
  for (int f = 0; f < 64; ++f) m = fmaxf(m, b[f * 130 + v]);
  float s = 0.f;
  for (int f = 0; f < 64; ++f) s += __expf((b[f * 130 + v] - m) * 100.0f);
  const float inv = 1.0f / s;

  float accv = 0.f;
  for (int fp = 0; fp < 64; ++fp) {
    float e = __expf((b[fp * 130 + v] - m) * 100.0f) * inv;
    float g = 0.f;
    for (int f = 0; f < 64; ++f) g += Fl[fp * 64 + f] * b[f * 130 + v];
    accv += e * g;
  }
  out[(long)t * 128 + v] = accv;
}

// ---------------------------------------------------------------------------
// Final loss: sim = (ts+vw+sf+fw)/4; loss = (CE(sim) + CE(sim^T)) / 2
// One block of 128 threads.
// ---------------------------------------------------------------------------
__global__ __launch_bounds__(128) void loss_kernel(
    const float* __restrict__ ts, const float* __restrict__ vw,
    const float* __restrict__ sf, const float* __restrict__ fw,
    float* __restrict__ out)
{
  __shared__ float red[128];
  const int i = threadIdx.x;

  auto sim = [&](int r, int c) -> float {
    long k = (long)r * 128 + c;
    return 0.25f * (ts[k] + vw[k] + sf[k] + fw[k]);
  };

  // row i log-softmax diag term
  float m = -3.4e38f;
  for (int j = 0; j < 128; ++j) m = fmaxf(m, sim(i, j));
  float s = 0.f;
  for (int j = 0; j < 128; ++j) s += __expf(sim(i, j) - m);
  float rowterm = sim(i, i) - m - __logf(s);

  // column i log-softmax diag term
  float m2 = -3.4e38f;
  for (int j = 0; j < 128; ++j) m2 = fmaxf(m2, sim(j, i));
  float s2 = 0.f;
  for (int j = 0; j < 128; ++j) s2 += __expf(sim(j, i) - m2);
  float colterm = sim(i, i) - m2 - __logf(s2);

  red[i] = rowterm + colterm;
  __syncthreads();
  if (i == 0) {
    float tot = 0.f;
    for (int j = 0; j < 128; ++j) tot += red[j];
    out[0] = -tot / 256.0f;   // -(mean_row + mean_col)/2
  }
}

// ---------------------------------------------------------------------------
extern "C" void kernel_launch(void* const* d_in, const int* in_sizes, int n_in,
                              void* d_out, int out_size, void* d_ws, size_t ws_size,
                              hipStream_t stream)
{
  const float* traj   = (const float*)d_in[0];   // [128,768]
  const float* frame  = (const float*)d_in[1];   // [128,64,768]
  const float* sent   = (const float*)d_in[2];   // [128,768]
  const float* word   = (const float*)d_in[3];   // [128,80,768]
  const float* Gm     = (const float*)d_in[4];   // [768,768]
  const float* Wlogit = (const float*)d_in[5];   // [80,80]
  const float* Flogit = (const float*)d_in[6];   // [64,64]
  const float* Lm     = (const float*)d_in[7];   // [768,768]
  const float* Fm     = (const float*)d_in[8];   // [64,64]
  const float* Wm     = (const float*)d_in[9];   // [80,80]
  const float* Fm2    = (const float*)d_in[10];  // [64,64]
  const float* Wm2    = (const float*)d_in[11];  // [80,80]

  float* ws = (float*)d_ws;
  float* WL = ws;                          // 10240*768 = 7,864,320
  float* Am = WL + (size_t)10240 * 768;    // 10240*128 = 1,310,720
  float* Bm = Am + (size_t)10240 * 128;    // 128*8192  = 1,048,576
  float* sg = Bm + (size_t)128 * 8192;     // 128*768   =    98,304
  float* ts = sg + (size_t)128 * 768;      // 128*128
  float* vw = ts + 16384;
  float* sf = vw + 16384;
  float* fw = sf + 16384;

  // 1) WL = word @ local_mat           [10240,768] x [768,768]
  gemm_nn_kernel<<<dim3(10240 / 32, 768 / 64), 32, 0, stream>>>(word, Lm, WL, 768, 768);
  // 2) sg = sentence @ global_mat      [128,768] x [768,768]
  gemm_nn_kernel<<<dim3(128 / 32, 768 / 64), 32, 0, stream>>>(sent, Gm, sg, 768, 768);
  // 3) ts = sg @ traj^T                [128,768] x [128,768]^T
  gemm_nt_kernel<<<dim3(128 / 32, 128 / 64), 32, 0, stream>>>(sg, traj, ts, 768, 128);
  // 4) Am = word @ traj^T              [10240,768] x [128,768]^T
  gemm_nt_kernel<<<dim3(10240 / 32, 128 / 64), 32, 0, stream>>>(word, traj, Am, 768, 128);
  // 5) Bm = sentence @ frame^T         [128,768] x [8192,768]^T
  gemm_nt_kernel<<<dim3(128 / 32, 8192 / 64), 32, 0, stream>>>(sent, frame, Bm, 768, 8192);
  // 6) video_word_sim
  reduce_video_word_kernel<<<128, 128, 0, stream>>>(Am, Wlogit, vw);
  // 7) sentence_frame_sim
  reduce_sentence_frame_kernel<<<128, 128, 0, stream>>>(Bm, Flogit, sf);
  // 8) frame_word_sim (fused S GEMM + async-LDS staged epilogue)
  frame_word_kernel<<<dim3(128, 128), 160, 0, stream>>>(WL, frame, Wm, Fm, Fm2, Wm2, fw);
  // 9) final double cross-entropy -> scalar
  loss_kernel<<<1, 128, 0, stream>>>(ts, vw, sf, fw, (float*)d_out);
}